// Encoder_678604833557
// MI455X (gfx1250) — compile-verified
//
#include <hip/hip_runtime.h>
#include <hip/hip_bf16.h>

// ---------------- problem constants (match reference) ----------------
#define NN 50000
#define EE 800000
#define GG 500
#define DD 64
#define BN_EPS 1e-5f
#define ZS_STRIDE 68   // LDS row stride (floats): conflict-free b64 access

typedef float v2f __attribute__((ext_vector_type(2)));
typedef float v8f __attribute__((ext_vector_type(8)));

// ---------------- generic zero fill ----------------
__global__ void k_zero(float* __restrict__ p, int n) {
    int i = blockIdx.x * blockDim.x + threadIdx.x;
    if (i < n) p[i] = 0.0f;
}

// ---------------- weight repack into WMMA B-fragment order ----------------
// For s = 0..31 (s = (k0/4)*2 + half) and n = 0..63:
//   Wp[(s*64+n)*2 + {0,1}] = W[(4*(s>>1) + 2*(s&1) + {0,1})*64 + n]
// In the GEMM, lane (half, n) at step k0 loads the b64 at
//   Wp + (half*64 + n)*2 + k0*64     (constant 1 KB stride across unrolled loop)
__global__ void k_pack_w(const float* __restrict__ W, float* __restrict__ Wp) {
    int t = blockIdx.x * 256 + threadIdx.x;  // 0..2047
    if (t >= 32 * 64) return;
    int s = t >> 6;
    int n = t & 63;
    int ka = 4 * (s >> 1) + 2 * (s & 1);
    Wp[t * 2 + 0] = W[ka * 64 + n];
    Wp[t * 2 + 1] = W[(ka + 1) * 64 + n];
}

// ---------------- edge scatter-add: agg[dst] += h[src] ----------------
// 16 threads per edge, 4 floats per thread (float4 gather, 4 fp32 atomics)
__global__ void k_edge_scatter(const int* __restrict__ src, const int* __restrict__ dst,
                               const float* __restrict__ h, float* __restrict__ agg) {
    int t = blockIdx.x * 256 + threadIdx.x;
    int e = t >> 4;
    if (e >= EE) return;
    int f = (t & 15) * 4;
    int s = src[e];
    int d = dst[e];
    const float4 v = *(const float4*)(h + (size_t)s * DD + f);
    float* p = agg + (size_t)d * DD + f;
    atomicAdd(p + 0, v.x);
    atomicAdd(p + 1, v.y);
    atomicAdd(p + 2, v.z);
    atomicAdd(p + 3, v.w);
}

// ---------------- fused GIN MLP: out = relu( relu((in+agg)@W1+b1) @ W2 + b2 ) ----
// Block: 128 threads (4 waves). Block handles 16 rows; wave w handles cols [16w,16w+16).
// V_WMMA_F32_16X16X4_F32 fragments:
//   A (16x4): lane -> m = lane&15, half = lane>>4; elems = A[m][k0+2*half + {0,1}]
//   B (4x16): lane -> n; elems = B[k0+2*half + {0,1}][n]  (prepacked, single b64)
//   C/D (16x16): elem r of lane -> row 8*half + r, col lane&15
// Requirement: `in`/`agg` must be readable for ceil16(nrows) rows (caller pads).
template <bool HAS_AGG, bool GUARD>
__global__ void k_gin_mlp(const float* __restrict__ in, const float* __restrict__ agg,
                          const float* __restrict__ W1p, const float* __restrict__ b1,
                          const float* __restrict__ W2p, const float* __restrict__ b2,
                          float* __restrict__ out, int nrows) {
    __shared__ float zs[16 * ZS_STRIDE];

    const int wave = threadIdx.x >> 5;   // 0..3
    const int lane = threadIdx.x & 31;
    const int half = lane >> 4;
    const int lm   = lane & 15;
    const int n    = wave * 16 + lm;     // output column owned by this lane
    const int row0 = blockIdx.x * 16;
    const int m    = row0 + lm;

    // -------- stage 1: Z = relu((in+agg) @ W1 + b1) --------
    v8f acc = {};
    {
        const float* arow = in + (size_t)m * DD + 2 * half;
        const float* grow = HAS_AGG ? (agg + (size_t)m * DD + 2 * half) : nullptr;
        const float* bcol = W1p + ((size_t)half * 64 + n) * 2;
        #pragma unroll
        for (int k0 = 0; k0 < DD; k0 += 4) {
            v2f a = *(const v2f*)(arow + k0);
            if constexpr (HAS_AGG) {
                v2f g = *(const v2f*)(grow + k0);
                a.x += g.x;
                a.y += g.y;
            }
            v2f b = *(const v2f*)(bcol + (size_t)k0 * 64);
            acc = __builtin_amdgcn_wmma_f32_16x16x4_f32(false, a, false, b, (short)0, acc,
                                                        false, false);
        }
    }
    {
        const float bias = b1[n];
        #pragma unroll
        for (int r = 0; r < 8; ++r) {
            float v = acc[r] + bias;
            v = v > 0.0f ? v : 0.0f;
            zs[(half * 8 + r) * ZS_STRIDE + n] = v;
        }
    }
    __syncthreads();

    // -------- stage 2: Y = relu(Z @ W2 + b2) --------
    v8f acc2 = {};
    {
        const float* zrow = &zs[lm * ZS_STRIDE + 2 * half];
        const float* bcol = W2p + ((size_t)half * 64 + n) * 2;
        #pragma unroll
        for (int k0 = 0; k0 < DD; k0 += 4) {
            v2f a = *(const v2f*)(zrow + k0);
            v2f b = *(const v2f*)(bcol + (size_t)k0 * 64);
            acc2 = __builtin_amdgcn_wmma_f32_16x16x4_f32(false, a, false, b, (short)0, acc2,
                                                         false, false);
        }
    }
    {
        const float bias = b2[n];
        #pragma unroll
        for (int r = 0; r < 8; ++r) {
            const int mr = row0 + half * 8 + r;
            float v = acc2[r] + bias;
            v = v > 0.0f ? v : 0.0f;
            if constexpr (GUARD) {
                if (mr < nrows) out[(size_t)mr * DD + n] = v;
            } else {
                out[(size_t)mr * DD + n] = v;
            }
        }
    }
}

// ---------------- per-column sum / sumsq over nrows x 64 ----------------
__global__ void k_col_stats(const float* __restrict__ a, float* __restrict__ stats, int nrows) {
    const int col = threadIdx.x & 63;
    const int grp = threadIdx.x >> 6;  // 0..3
    float s = 0.0f, q = 0.0f;
    for (int r = blockIdx.x * 4 + grp; r < nrows; r += gridDim.x * 4) {
        float v = a[(size_t)r * DD + col];
        s += v;
        q += v * v;
    }
    __shared__ float ls[256];
    __shared__ float lq[256];
    ls[threadIdx.x] = s;
    lq[threadIdx.x] = q;
    __syncthreads();
    if (threadIdx.x < 64) {
        s = ls[threadIdx.x] + ls[threadIdx.x + 64] + ls[threadIdx.x + 128] + ls[threadIdx.x + 192];
        q = lq[threadIdx.x] + lq[threadIdx.x + 64] + lq[threadIdx.x + 128] + lq[threadIdx.x + 192];
        atomicAdd(&stats[col], s);
        atomicAdd(&stats[64 + col], q);
    }
}

// ---------------- BN finalize: fold (mean,var,gamma,beta) -> (scale,shift) ----------
__global__ void k_bn_finalize(float* __restrict__ stats, const float* __restrict__ gamma,
                              const float* __restrict__ beta, int nrows) {
    const int c = threadIdx.x;
    if (c >= 64) return;
    const float inv_n = 1.0f / (float)nrows;
    const float mean = stats[c] * inv_n;
    float var = stats[64 + c] * inv_n - mean * mean;
    const float scale = gamma[c] * rsqrtf(var + BN_EPS);
    stats[c] = scale;
    stats[64 + c] = beta[c] - mean * scale;
}

// ---------------- BN apply: out = a*scale + shift ----------------
__global__ void k_bn_apply(const float* __restrict__ a, const float* __restrict__ stats,
                           float* __restrict__ out, int n) {
    int i = blockIdx.x * blockDim.x + threadIdx.x;
    if (i >= n) return;
    const int c = i & 63;
    out[i] = a[i] * stats[c] + stats[64 + c];
}

// ---------------- global mean pool: accumulate ----------------
__global__ void k_pool_accum(const float* __restrict__ h, const int* __restrict__ batch,
                             float* __restrict__ pool, float* __restrict__ cnt) {
    int t = blockIdx.x * 256 + threadIdx.x;
    if (t >= NN * DD) return;
    const int i = t >> 6;
    const int c = t & 63;
    const int g = batch[i];
    atomicAdd(&pool[(size_t)g * DD + c], h[t]);
    if (c == 0) atomicAdd(&cnt[g], 1.0f);
}

__global__ void k_pool_div(const float* __restrict__ pool, const float* __restrict__ cnt,
                           float* __restrict__ pooled) {
    int t = blockIdx.x * 256 + threadIdx.x;
    if (t >= GG * DD) return;
    const int g = t >> 6;
    float c = cnt[g];
    c = c > 1.0f ? c : 1.0f;
    pooled[t] = pool[t] / c;
}

// =====================================================================
extern "C" void kernel_launch(void* const* d_in, const int* in_sizes, int n_in,
                              void* d_out, int out_size, void* d_ws, size_t ws_size,
                              hipStream_t stream) {
    (void)in_sizes; (void)n_in; (void)out_size; (void)ws_size;

    // ---- inputs (setup_inputs dict order, nested lists flattened) ----
    const float* x     = (const float*)d_in[0];
    const int*   ei    = (const int*)d_in[1];
    const int*   src   = ei;
    const int*   dst   = ei + EE;
    const int*   batch = (const int*)d_in[2];
    const float* cW1[5]; const float* cb1[5]; const float* cW2[5]; const float* cb2[5];
    const float* bng[5]; const float* bnb[5];
    for (int i = 0; i < 5; ++i) {
        cW1[i] = (const float*)d_in[3 + 4 * i + 0];
        cb1[i] = (const float*)d_in[3 + 4 * i + 1];
        cW2[i] = (const float*)d_in[3 + 4 * i + 2];
        cb2[i] = (const float*)d_in[3 + 4 * i + 3];
        bng[i] = (const float*)d_in[23 + 2 * i + 0];
        bnb[i] = (const float*)d_in[23 + 2 * i + 1];
    }
    const float* muW1 = (const float*)d_in[33];
    const float* mub1 = (const float*)d_in[34];
    const float* muW2 = (const float*)d_in[35];
    const float* mub2 = (const float*)d_in[36];
    const float* lvW1 = (const float*)d_in[37];
    const float* lvb1 = (const float*)d_in[38];
    const float* lvW2 = (const float*)d_in[39];
    const float* lvb2 = (const float*)d_in[40];

    float* out = (float*)d_out;
    float* out_node_mu = out;                       // N*D
    float* out_node_lv = out + (size_t)NN * DD;     // N*D
    float* out_cls_mu  = out + (size_t)2 * NN * DD; // G*D
    float* out_cls_lv  = out_cls_mu + (size_t)GG * DD;

    // ---- workspace layout (floats) ----
    float* ws     = (float*)d_ws;
    float* hbuf   = ws;                             // N*D : current activations
    float* bbuf   = hbuf + (size_t)NN * DD;         // N*D : agg, then gemm output (in-place safe)
    float* stats  = bbuf + (size_t)NN * DD;         // 128 : sum/sumsq -> scale/shift
    float* pool   = stats + 128;                    // G*D
    float* cnt    = pool + (size_t)GG * DD;         // G
    float* pooled = cnt + GG;                       // 512*D (padded with zero rows for WMMA)
    float* wpack  = pooled + (size_t)512 * DD;      // 14 * 4096 : repacked weights

    float* pW1[5]; float* pW2[5];
    for (int i = 0; i < 5; ++i) {
        pW1[i] = wpack + (size_t)(2 * i + 0) * 4096;
        pW2[i] = wpack + (size_t)(2 * i + 1) * 4096;
    }
    float* pmuW1 = wpack + (size_t)10 * 4096;
    float* pmuW2 = wpack + (size_t)11 * 4096;
    float* plvW1 = wpack + (size_t)12 * 4096;
    float* plvW2 = wpack + (size_t)13 * 4096;

    const int ND = NN * DD;
    const dim3 b256(256), b128(128), b64(64);
    const int gemm_blocks_N = NN / 16;              // 3125 (exact)
    const int gemm_blocks_G = (GG + 15) / 16;       // 32
    const int scat_blocks = (EE * 16 + 255) / 256;  // 50000

    // ---- one-shot weight repacks (deterministic every call) ----
    for (int i = 0; i < 5; ++i) {
        k_pack_w<<<8, b256, 0, stream>>>(cW1[i], pW1[i]);
        k_pack_w<<<8, b256, 0, stream>>>(cW2[i], pW2[i]);
    }
    k_pack_w<<<8, b256, 0, stream>>>(muW1, pmuW1);
    k_pack_w<<<8, b256, 0, stream>>>(muW2, pmuW2);
    k_pack_w<<<8, b256, 0, stream>>>(lvW1, plvW1);
    k_pack_w<<<8, b256, 0, stream>>>(lvW2, plvW2);

    // ---- 3 trunk GIN layers ----
    const float* cur = x;
    for (int i = 0; i < 3; ++i) {
        k_zero<<<(ND + 255) / 256, b256, 0, stream>>>(bbuf, ND);
        k_edge_scatter<<<scat_blocks, b256, 0, stream>>>(src, dst, cur, bbuf);
        k_gin_mlp<true, false><<<gemm_blocks_N, b128, 0, stream>>>(
            cur, bbuf, pW1[i], cb1[i], pW2[i], cb2[i], bbuf, NN);
        k_zero<<<1, b256, 0, stream>>>(stats, 128);
        k_col_stats<<<512, b256, 0, stream>>>(bbuf, stats, NN);
        k_bn_finalize<<<1, b64, 0, stream>>>(stats, bng[i], bnb[i], NN);
        k_bn_apply<<<(ND + 255) / 256, b256, 0, stream>>>(bbuf, stats, hbuf, ND);
        cur = hbuf;
    }

    // ---- global mean pool over final trunk activations (pooled padded to 512 rows) ----
    k_zero<<<(GG * DD + GG + 255) / 256, b256, 0, stream>>>(pool, GG * DD + GG);
    k_zero<<<(512 * DD + 255) / 256, b256, 0, stream>>>(pooled, 512 * DD);
    k_pool_accum<<<(ND + 255) / 256, b256, 0, stream>>>(hbuf, batch, pool, cnt);
    k_pool_div<<<(GG * DD + 255) / 256, b256, 0, stream>>>(pool, cnt, pooled);

    // ---- node heads: mu (conv_params[3], bn[3]) and logvar (conv_params[4], bn[4]) ----
    float* head_out[2] = { out_node_mu, out_node_lv };
    for (int hidx = 0; hidx < 2; ++hidx) {
        const int p = 3 + hidx;
        k_zero<<<(ND + 255) / 256, b256, 0, stream>>>(bbuf, ND);
        k_edge_scatter<<<scat_blocks, b256, 0, stream>>>(src, dst, hbuf, bbuf);
        k_gin_mlp<true, false><<<gemm_blocks_N, b128, 0, stream>>>(
            hbuf, bbuf, pW1[p], cb1[p], pW2[p], cb2[p], bbuf, NN);
        k_zero<<<1, b256, 0, stream>>>(stats, 128);
        k_col_stats<<<512, b256, 0, stream>>>(bbuf, stats, NN);
        k_bn_finalize<<<1, b64, 0, stream>>>(stats, bng[p], bnb[p], NN);
        k_bn_apply<<<(ND + 255) / 256, b256, 0, stream>>>(bbuf, stats, head_out[hidx], ND);
    }

    // ---- class heads: relu(mlp(pooled)) ----
    k_gin_mlp<false, true><<<gemm_blocks_G, b128, 0, stream>>>(
        pooled, (const float*)nullptr, pmuW1, mub1, pmuW2, mub2, out_cls_mu, GG);
    k_gin_mlp<false, true><<<gemm_blocks_G, b128, 0, stream>>>(
        pooled, (const float*)nullptr, plvW1, lvb1, plvW2, lvb2, out_cls_lv, GG);
}